// SpectralConv_25099788878476
// MI455X (gfx1250) — compile-verified
//
#include <hip/hip_runtime.h>
#include <hip/hip_bf16.h>

// FNO 2D spectral conv, decomposed into mode-truncated DFT GEMMs on f32 WMMA.
// B=8, H=256, W=256, CIN=COUT=32, M1=M2=16.
//
// Workspace layout (floats):
//   tab   [49152]              twiddle tables: F1(8192) C2(8192) S2(8192) IC(8192) IS(8192) P(8192)
//   Y     [B][H][2][16][32]    = 2,097,152   stage-1 output (W-dir spectrum)
//   Z     [B][2][32][16][32]   =   262,144   stage-2 output (2D mode block)
//   U     [B][2][32][16][32]   =   262,144   stage-3 output (channel-mixed modes)
//   T     [B][H][2][16][32]    = 2,097,152   stage-4 output (H-dir inverse)
// total ~19.1 MB  (fits easily in the 192 MB L2 -> only x-read + out-write hit HBM)

typedef float v2f __attribute__((ext_vector_type(2)));
typedef float v8f __attribute__((ext_vector_type(8)));

#define WMMA_F32(a, b, c) \
  __builtin_amdgcn_wmma_f32_16x16x4_f32(false, (a), false, (b), (short)0, (c), false, false)

static constexpr int Bn = 8, Hn = 256, Wn = 256, Cn = 32;
static constexpr int TAB_F1 = 0;
static constexpr int TAB_C2 = 8192;
static constexpr int TAB_S2 = 16384;
static constexpr int TAB_IC = 24576;
static constexpr int TAB_IS = 32768;
static constexpr int TAB_P  = 40960;
static constexpr int TAB_SZ = 49152;

// ---------------------------------------------------------------- tables ----
__global__ void fno_init_tables(float* __restrict__ tab) {
  int t = blockIdx.x * blockDim.x + threadIdx.x;
  if (t >= 32 * 256) return;
  int j = t >> 8;    // 0..31
  int n = t & 255;   // 0..255
  const float STEP = 6.28318530717958647692f / 256.0f;

  // F1: forward W-DFT A-matrix [32][256]; rows 0..15 = cos, 16..31 = -sin
  int kwj = j & 15;
  float s1, c1;
  __sincosf((float)((kwj * n) & 255) * STEP, &s1, &c1);
  tab[TAB_F1 + j * 256 + n] = (j < 16) ? c1 : -s1;

  // C2/S2: forward H-DFT [32][256]; row j -> freq kh (low block 0..15, high 240..255)
  int kh = (j < 16) ? j : (240 + (j - 16));
  float s2, c2;
  __sincosf((float)((kh * n) & 255) * STEP, &s2, &c2);
  tab[TAB_C2 + j * 256 + n] = c2;
  tab[TAB_S2 + j * 256 + n] = s2;

  // IC/IS: inverse H-DFT [256][32] (h-major), 1/H folded in
  tab[TAB_IC + n * 32 + j] = c2 * (1.0f / 256.0f);
  tab[TAB_IS + n * 32 + j] = s2 * (1.0f / 256.0f);

  // P: inverse W (irfft) matrix [256][32]; cols 0..15: c_k*cos/W, 16..31: -c_k*sin/W
  // c_0 = 1 (and its imag column = 0, matching irfft discarding Im of DC bin), else 2.
  float pv;
  if (j < 16) {
    pv = ((j == 0) ? 1.0f : 2.0f) * c1 * (1.0f / 256.0f);
  } else {
    int k = j - 16;
    pv = (k == 0) ? 0.0f : (-2.0f * s1 * (1.0f / 256.0f));
  }
  tab[TAB_P + n * 32 + j] = pv;
}

// ------------------------------------------------- stage 1: W-dir forward ---
// per (b,h): Y[32 j][32 c] = F1[32][256] x Xslab[256 w][32 c]   (contiguous slab)
__global__ void fno_stage1(const float* __restrict__ x, const float* __restrict__ tab,
                           float* __restrict__ Y) {
  const float* F1 = tab + TAB_F1;
  int bh   = blockIdx.x;                    // 0..2047
  int wave = threadIdx.x >> 5;              // 4 waves: tile (mi,ni)
  int mi = wave >> 1, ni = wave & 1;
  int lane = threadIdx.x & 31;
  int half = lane >> 4, lr = lane & 15, koff = half * 2;
  const float* X = x + (size_t)bh * (Wn * Cn);
  int m = mi * 16 + lr;                     // A row (j)
  int n = ni * 16 + lr;                     // B col (c)
  v8f acc = {};
  for (int k0 = 0; k0 < 256; k0 += 4) {
    int k = k0 + koff;
    v2f a, b;
    a.x = F1[m * 256 + k];      a.y = F1[m * 256 + k + 1];
    b.x = X[k * Cn + n];        b.y = X[(k + 1) * Cn + n];
    acc = WMMA_F32(a, b, acc);
  }
  float* Yb = Y + (size_t)bh * 1024;        // [2][16][32] == [j(32)][c(32)]
#pragma unroll
  for (int r = 0; r < 8; ++r) {
    int row = mi * 16 + r + 8 * half;
    Yb[row * 32 + ni * 16 + lr] = acc[r];
  }
}

// ------------------------------------------------- stage 2: H-dir forward ---
// per (b,kw): Z = G x Y (complex, contract over h=256)
//   Zre = C2*Yre + S2*Yim ; Zim = C2*Yim - S2*Yre
__global__ void fno_stage2(const float* __restrict__ tab, const float* __restrict__ Y,
                           float* __restrict__ Z) {
  const float* C2 = tab + TAB_C2;
  const float* S2 = tab + TAB_S2;
  int b  = blockIdx.x >> 4;
  int kw = blockIdx.x & 15;
  int wave = threadIdx.x >> 5;
  int mi = wave >> 1, ni = wave & 1;
  int lane = threadIdx.x & 31;
  int half = lane >> 4, lr = lane & 15, koff = half * 2;
  int m = mi * 16 + lr;
  int n = ni * 16 + lr;
  const float* Yb = Y + (size_t)b * (Hn * 1024);   // [h][2][16][32]
  v8f accRe = {}, accIm = {};
  for (int k0 = 0; k0 < 256; k0 += 4) {
    int h = k0 + koff;
    v2f aC, aS, aSn, bRe, bIm;
    aC.x = C2[m * 256 + h];  aC.y = C2[m * 256 + h + 1];
    aS.x = S2[m * 256 + h];  aS.y = S2[m * 256 + h + 1];
    aSn.x = -aS.x;           aSn.y = -aS.y;           // f32 WMMA has no A-negate
    bRe.x = Yb[(size_t)h * 1024 + kw * 32 + n];
    bRe.y = Yb[(size_t)(h + 1) * 1024 + kw * 32 + n];
    bIm.x = Yb[(size_t)h * 1024 + 512 + kw * 32 + n];
    bIm.y = Yb[(size_t)(h + 1) * 1024 + 512 + kw * 32 + n];
    accRe = WMMA_F32(aC,  bRe, accRe);
    accRe = WMMA_F32(aS,  bIm, accRe);
    accIm = WMMA_F32(aC,  bIm, accIm);
    accIm = WMMA_F32(aSn, bRe, accIm);
  }
#pragma unroll
  for (int r = 0; r < 8; ++r) {
    int khid = mi * 16 + r + 8 * half;
    int c    = ni * 16 + lr;
    Z[((((size_t)b * 2 + 0) * 32 + khid) * 16 + kw) * 32 + c] = accRe[r];
    Z[((((size_t)b * 2 + 1) * 32 + khid) * 16 + kw) * 32 + c] = accIm[r];
  }
}

// --------------------------------------------- stage 3: channel mix (VALU) --
// per (b,khid,kw,o): complex dot over i=32 with per-mode weight matrix
__global__ void fno_stage3(const float* __restrict__ Z, const float* __restrict__ wr,
                           const float* __restrict__ wi, float* __restrict__ U) {
  int t = blockIdx.x * blockDim.x + threadIdx.x;   // 131072 threads
  int o    = t & 31;
  int kw   = (t >> 5) & 15;
  int khid = (t >> 9) & 31;
  int b    = t >> 14;
  int blk  = khid >> 4;       // weight block (low/high freq)
  int m1   = khid & 15;
  const float* zr = Z + ((((size_t)b * 2 + 0) * 32 + khid) * 16 + kw) * 32;
  const float* zi = Z + ((((size_t)b * 2 + 1) * 32 + khid) * 16 + kw) * 32;
  float ure = 0.0f, uim = 0.0f;
#pragma unroll 8
  for (int i = 0; i < 32; ++i) {
    size_t widx = ((((size_t)blk * 32 + i) * 32 + o) * 16 + m1) * 16 + kw;
    float WR = wr[widx], WI = wi[widx];
    float zre = zr[i],   zim = zi[i];
    ure += zre * WR - zim * WI;
    uim += zre * WI + zim * WR;
  }
  U[((((size_t)b * 2 + 0) * 32 + khid) * 16 + kw) * 32 + o] = ure;
  U[((((size_t)b * 2 + 1) * 32 + khid) * 16 + kw) * 32 + o] = uim;
}

// ------------------------------------------------- stage 4: H-dir inverse ---
// per (b,kw): T[256 h][32 o] = Ginv x U (complex, contract over 32 kh modes)
//   Tre = IC*Ure - IS*Uim ; Tim = IC*Uim + IS*Ure   (1/H inside IC/IS)
__global__ void fno_stage4(const float* __restrict__ tab, const float* __restrict__ U,
                           float* __restrict__ T) {
  const float* IC = tab + TAB_IC;   // [256][32]
  const float* IS = tab + TAB_IS;
  int b  = blockIdx.x >> 4;
  int kw = blockIdx.x & 15;
  int wave = threadIdx.x >> 5;      // 8 waves x 4 tiles = 32 tiles (16 mtiles x 2 ntiles)
  int lane = threadIdx.x & 31;
  int half = lane >> 4, lr = lane & 15, koff = half * 2;
  for (int it = 0; it < 4; ++it) {
    int tile = wave * 4 + it;
    int mi = tile >> 1, ni = tile & 1;
    int m = mi * 16 + lr;           // h row
    int n = ni * 16 + lr;           // o col
    v8f accRe = {}, accIm = {};
    for (int k0 = 0; k0 < 32; k0 += 4) {
      int k = k0 + koff;
      v2f aC, aS, aSn, bRe, bIm;
      aC.x = IC[m * 32 + k];  aC.y = IC[m * 32 + k + 1];
      aS.x = IS[m * 32 + k];  aS.y = IS[m * 32 + k + 1];
      aSn.x = -aS.x;          aSn.y = -aS.y;
      bRe.x = U[((((size_t)b * 2 + 0) * 32 + k) * 16 + kw) * 32 + n];
      bRe.y = U[((((size_t)b * 2 + 0) * 32 + k + 1) * 16 + kw) * 32 + n];
      bIm.x = U[((((size_t)b * 2 + 1) * 32 + k) * 16 + kw) * 32 + n];
      bIm.y = U[((((size_t)b * 2 + 1) * 32 + k + 1) * 16 + kw) * 32 + n];
      accRe = WMMA_F32(aC,  bRe, accRe);
      accRe = WMMA_F32(aSn, bIm, accRe);
      accIm = WMMA_F32(aC,  bIm, accIm);
      accIm = WMMA_F32(aS,  bRe, accIm);
    }
#pragma unroll
    for (int r = 0; r < 8; ++r) {
      int h = mi * 16 + r + 8 * half;
      int o = ni * 16 + lr;
      T[(((size_t)b * Hn + h) * 2 + 0) * 512 + kw * 32 + o] = accRe[r];
      T[(((size_t)b * Hn + h) * 2 + 1) * 512 + kw * 32 + o] = accIm[r];
    }
  }
}

// ---------------------------------------- stage 5: W-dir inverse (irfft) ----
// per (b,h): out[256 w][32 o] = P[256][32] x Tslab[32 k][32 o]; k = reim*16+kw
__global__ void fno_stage5(const float* __restrict__ tab, const float* __restrict__ T,
                           float* __restrict__ out) {
  const float* P = tab + TAB_P;     // [256][32]
  int bh = blockIdx.x;              // 0..2047
  const float* Ts = T + (size_t)bh * 1024;     // contiguous [32][32]
  float* O = out + (size_t)bh * (Wn * Cn);     // contiguous [256][32], channels-last
  int wave = threadIdx.x >> 5;      // 8 waves x 4 tiles = 32 tiles
  int lane = threadIdx.x & 31;
  int half = lane >> 4, lr = lane & 15, koff = half * 2;
  for (int it = 0; it < 4; ++it) {
    int tile = wave * 4 + it;
    int mi = tile >> 1, ni = tile & 1;
    int m = mi * 16 + lr;           // w row
    int n = ni * 16 + lr;           // o col
    v8f acc = {};
    for (int k0 = 0; k0 < 32; k0 += 4) {
      int k = k0 + koff;
      v2f a, b;
      a.x = P[m * 32 + k];        a.y = P[m * 32 + k + 1];
      b.x = Ts[k * 32 + n];       b.y = Ts[(k + 1) * 32 + n];
      acc = WMMA_F32(a, b, acc);
    }
#pragma unroll
    for (int r = 0; r < 8; ++r) {
      int w = mi * 16 + r + 8 * half;
      O[(size_t)w * 32 + ni * 16 + lr] = acc[r];
    }
  }
}

// ---------------------------------------------------------------- launch ----
extern "C" void kernel_launch(void* const* d_in, const int* in_sizes, int n_in,
                              void* d_out, int out_size, void* d_ws, size_t ws_size,
                              hipStream_t stream) {
  const float* x  = (const float*)d_in[0];   // [8,256,256,32]
  const float* wr = (const float*)d_in[1];   // [2,32,32,16,16]
  const float* wi = (const float*)d_in[2];
  float* out = (float*)d_out;                // [8,256,256,32]

  float* tab = (float*)d_ws;
  float* Y = tab + TAB_SZ;                   // 2,097,152
  float* Z = Y + (size_t)Bn * Hn * 2 * 16 * 32;
  float* U = Z + (size_t)Bn * 2 * 32 * 16 * 32;
  float* T = U + (size_t)Bn * 2 * 32 * 16 * 32;

  fno_init_tables<<<32, 256, 0, stream>>>(tab);
  fno_stage1<<<Bn * Hn, 128, 0, stream>>>(x, tab, Y);        // 2048 blocks, 4 waves
  fno_stage2<<<Bn * 16, 128, 0, stream>>>(tab, Y, Z);        // 128 blocks
  fno_stage3<<<512, 256, 0, stream>>>(Z, wr, wi, U);         // 131072 threads
  fno_stage4<<<Bn * 16, 256, 0, stream>>>(tab, U, T);        // 128 blocks, 8 waves
  fno_stage5<<<Bn * Hn, 256, 0, stream>>>(tab, T, out);      // 2048 blocks, 8 waves
}